// GPSSMs_40905268527428
// MI455X (gfx1250) — compile-verified
//
#include <hip/hip_runtime.h>
#include <math.h>

#define B_   32
#define T_   16
#define DY   16
#define DX   32
#define M_   64
#define N_   50
#define PROC_SD 0.05f
#define EMIS_SD 0.1f
#define JITTER  1e-4f
#define LOC_R   5.0f

typedef __attribute__((ext_vector_type(2))) float v2f;
typedef __attribute__((ext_vector_type(8))) float v8f;

// ---- workspace layout (float offsets) ----
static constexpr size_t WS_TAPER = 0;                               // dx*dx
static constexpr size_t WS_LZZ   = WS_TAPER + (size_t)DX*DX;        // dx*M*M
static constexpr size_t WS_ALPHA = WS_LZZ   + (size_t)DX*M_*M_;     // N*dx*M
static constexpr size_t WS_XBUF  = WS_ALPHA + (size_t)N_*DX*M_;     // B*N*dx
static constexpr size_t WS_XM    = WS_XBUF  + (size_t)B_*N_*DX;     // B*dx
static constexpr size_t WS_LL    = WS_XM    + (size_t)B_*DX;        // T*B
static constexpr size_t WS_KL0   = WS_LL    + (size_t)T_*B_;        // 1
static constexpr size_t WS_GPKL  = WS_KL0 + 1;                      // dx

// ---------------- Gaspari-Cohn taper ----------------
__global__ void k_taper(float* ws){
  int i = blockIdx.x*blockDim.x + threadIdx.x;
  if (i < DX*DX){
    int r = i / DX, c = i % DX;
    int ad = (r > c) ? (r - c) : (c - r);
    int d  = (ad < DX - ad) ? ad : (DX - ad);
    float z = (float)d / LOC_R;
    float z2 = z*z, z3 = z2*z, z4 = z3*z, z5 = z4*z;
    float g = 0.f;
    if (z < 1.f)       g = 1.f - (5.f/3.f)*z2 + (5.f/8.f)*z3 + 0.5f*z4 - 0.25f*z5;
    else if (z < 2.f)  g = 4.f - 5.f*z + (5.f/3.f)*z2 + (5.f/8.f)*z3 - 0.5f*z4 + z5/12.f - 2.f/(3.f*z);
    ws[WS_TAPER + i] = g;
  }
}

// ---------------- KL[q(x0)||N(0,I)] ----------------
__global__ void k_kl0(const float* qm, const float* ql, float* ws){
  __shared__ float red[256];
  float s = 0.f;
  for (int i = threadIdx.x; i < B_*DX; i += 256){
    float l = ql[i], m = qm[i];
    s += expf(2.f*l) + m*m - 1.f - 2.f*l;
  }
  red[threadIdx.x] = s; __syncthreads();
  for (int st = 128; st > 0; st >>= 1){
    if (threadIdx.x < st) red[threadIdx.x] += red[threadIdx.x + st];
    __syncthreads();
  }
  if (threadIdx.x == 0) ws[WS_KL0] = 0.5f * red[0] / (float)B_;
}

// ---------------- Kzz build + Cholesky, one g per block (64 threads) ----------------
__global__ void k_kzz(const float* Z, const float* log_ls, const float* log_os, float* ws){
  __shared__ float Zl[M_][DX+1];
  __shared__ float z2l[M_];
  __shared__ float Kl[M_][M_+1];
  int g = blockIdx.x, m = threadIdx.x;
  float nhalf = -0.5f / expf(2.f*log_ls[g]);   // -0.5/ls2, hoisted reciprocal
  float osg   = expf(log_os[g]);
  float s2 = 0.f;
  for (int d = 0; d < DX; ++d){
    float v = Z[((size_t)g*M_ + m)*DX + d];
    Zl[m][d] = v; s2 += v*v;
  }
  z2l[m] = s2;
  __syncthreads();
  for (int j = 0; j < M_; ++j){
    float dot = 0.f;
    for (int d = 0; d < DX; ++d) dot += Zl[m][d]*Zl[j][d];
    float sq = z2l[m] + z2l[j] - 2.f*dot;
    Kl[m][j] = osg*expf(sq*nhalf) + ((m == j) ? JITTER : 0.f);
  }
  __syncthreads();
  // right-looking Cholesky (lower)
  for (int k = 0; k < M_; ++k){
    if (m == k) Kl[k][k] = sqrtf(Kl[k][k]);
    __syncthreads();
    if (m > k) Kl[m][k] = Kl[m][k] / Kl[k][k];
    __syncthreads();
    if (m > k){
      float f = Kl[m][k];
      for (int j = k+1; j <= m; ++j) Kl[m][j] -= f*Kl[j][k];
    }
    __syncthreads();
  }
  float* Lzz = ws + WS_LZZ + (size_t)g*M_*M_;
  for (int j = 0; j < M_; ++j) Lzz[m*M_ + j] = (j <= m) ? Kl[m][j] : 0.f;
}

// ---------------- U = m_u + Ls_t eps ; alpha = Kzz^{-1} U ----------------
__global__ void k_ualpha(const float* Ls, const float* m_u, const float* eps_U, float* ws){
  __shared__ float Lzzl[M_][M_+1];
  __shared__ float Lstl[M_][M_+1];
  __shared__ float invd[M_];
  __shared__ float eA[M_][M_];
  __shared__ float aA[M_][M_];
  int g = blockIdx.x, tid = threadIdx.x;
  const float* Lzz = ws + WS_LZZ + (size_t)g*M_*M_;
  for (int j = 0; j < M_; ++j){
    Lzzl[tid][j] = Lzz[tid*M_ + j];
    Lstl[tid][j] = (j <= tid) ? Ls[((size_t)g*M_ + tid)*M_ + j] : 0.f;
  }
  invd[tid] = 1.f / Lzz[tid*M_ + tid];
  __syncthreads();
  int n = tid;
  if (n < N_){
    for (int k = 0; k < M_; ++k) eA[n][k] = eps_U[((size_t)n*DX + g)*M_ + k];
    // u_i then forward solve Lzz a = u
    for (int i = 0; i < M_; ++i){
      float u = m_u[g*M_ + i];
      for (int k = 0; k <= i; ++k) u += Lstl[i][k]*eA[n][k];
      float a = u;
      for (int k = 0; k < i; ++k) a -= Lzzl[i][k]*aA[n][k];
      aA[n][i] = a * invd[i];
    }
    // backward solve Lzz^T alpha = a  (reuse eA as alpha)
    for (int j = M_-1; j >= 0; --j){
      float a = aA[n][j];
      for (int k = j+1; k < M_; ++k) a -= Lzzl[k][j]*eA[n][k];
      eA[n][j] = a * invd[j];
    }
    float* al = ws + WS_ALPHA + ((size_t)n*DX + g)*M_;
    for (int j = 0; j < M_; ++j) al[j] = eA[n][j];
  }
}

// ---------------- KL[q(u)||p(u)] per-g partial ----------------
__global__ void k_gpkl(const float* Ls, const float* m_u, float* ws){
  __shared__ float Lzzl[M_][M_+1];
  __shared__ float Lstl[M_][M_+1];
  __shared__ float invd[M_];
  __shared__ float vA[M_][M_];
  __shared__ float red[M_];
  int g = blockIdx.x, tid = threadIdx.x;
  const float* Lzz = ws + WS_LZZ + (size_t)g*M_*M_;
  for (int j = 0; j < M_; ++j){
    Lzzl[tid][j] = Lzz[tid*M_ + j];
    Lstl[tid][j] = (j <= tid) ? Ls[((size_t)g*M_ + tid)*M_ + j] : 0.f;
  }
  invd[tid] = 1.f / Lzz[tid*M_ + tid];
  __syncthreads();
  // V = Lzz^{-1} Ls_t  (thread tid owns column tid)
  float sv2 = 0.f;
  for (int i = 0; i < M_; ++i){
    float s = Lstl[i][tid];
    for (int k = 0; k < i; ++k) s -= Lzzl[i][k]*vA[k][tid];
    float v = s * invd[i];
    vA[i][tid] = v;
    sv2 += v*v;
  }
  red[tid] = sv2;
  __syncthreads();
  if (tid == 0){
    float tot = 0.f;
    for (int j = 0; j < M_; ++j) tot += red[j];
    float ldK = 0.f, ldS = 0.f;
    for (int i = 0; i < M_; ++i){
      ldK += logf(Lzzl[i][i]);
      ldS += logf(fabsf(Lstl[i][i]));
    }
    // w = Lzz^{-1} m_u (reuse vA column 0)
    float sw2 = 0.f;
    for (int i = 0; i < M_; ++i){
      float s = m_u[g*M_ + i];
      for (int k = 0; k < i; ++k) s -= Lzzl[i][k]*vA[k][0];
      float v = s * invd[i];
      vA[i][0] = v;
      sw2 += v*v;
    }
    ws[WS_GPKL + g] = tot + sw2 - (float)M_ + 2.f*ldK - 2.f*ldS;
  }
}

// ---------------- GP predict: x_t[n,g,b] via f32 WMMA, writes X[b][n][g] ----------------
// grid: (25, dx)  block: 128 (4 waves); wave tau = bx*4+w handles rows r0=16*tau of [N*B,dx]
__global__ void k_predict(int t, const float* Z, const float* log_ls, const float* log_os,
                          const float* qm, const float* ql, const float* eps_x0,
                          const float* eps_proc, float* ws){
  __shared__ float Zt[DX][M_+2];        // [k][m] = Z[g][m][k]
  __shared__ float z2l[M_];
  __shared__ float alphal[2][M_];
  __shared__ float xt[4][16][DX+1];
  __shared__ float x2l[4][16];
  int g    = blockIdx.y;
  int w    = threadIdx.x >> 5;
  int lane = threadIdx.x & 31;
  int tau  = blockIdx.x*4 + w;
  int n    = tau >> 1;                  // constant over the 16-row tile
  int b0   = (tau & 1) * 16;
  // cooperative: Z^T tile for this g
  for (int i = threadIdx.x; i < DX*M_; i += 128){
    int k = i >> 6, m = i & 63;
    Zt[k][m] = Z[((size_t)g*M_ + m)*DX + k];
  }
  // alpha rows for the two n's in this block
  for (int i = threadIdx.x; i < 2*M_; i += 128){
    int j = i >> 6, m = i & 63;
    alphal[j][m] = ws[WS_ALPHA + ((size_t)(2*blockIdx.x + j)*DX + g)*M_ + m];
  }
  // x tile for this wave (16 rows x dx)
  const float* xm = ws + WS_XM;
  for (int row = 0; row < 16; ++row){
    int b = b0 + row;
    float v;
    if (t == 0){
      float e = eps_x0[(((size_t)n*DX + g)*B_ + b)*DX + lane];
      v = qm[b*DX + lane] + expf(ql[b*DX + lane])*e;
    } else {
      v = xm[b*DX + lane];
    }
    xt[w][row][lane] = v;
  }
  __syncthreads();
  if (threadIdx.x < M_){
    float s = 0.f;
    for (int k = 0; k < DX; ++k){ float z = Zt[k][threadIdx.x]; s += z*z; }
    z2l[threadIdx.x] = s;
  }
  if (lane < 16){
    float s = 0.f;
    for (int k = 0; k < DX; ++k){ float x = xt[w][lane][k]; s += x*x; }
    x2l[w][lane] = s;
  }
  __syncthreads();
  float nhalf = -0.5f / expf(2.f*log_ls[g]);   // multiply, no per-element division
  float osg   = expf(log_os[g]);
  int kh = (lane < 16) ? 0 : 2;
  int c  = lane & 15;
  int aj = w >> 1;
  // preload A fragments (row = c, K = 32) once
  v2f afr[8];
  #pragma unroll
  for (int kc = 0; kc < 8; ++kc){
    int k0 = kc*4 + kh;
    afr[kc].x = xt[w][c][k0];
    afr[kc].y = xt[w][c][k0+1];
  }
  float pacc[8];
  #pragma unroll
  for (int i = 0; i < 8; ++i) pacc[i] = 0.f;
  // xz = x @ Z^T via V_WMMA_F32_16X16X4_F32, K=32 => 8 chunks; 4 column tiles of m
  for (int ct = 0; ct < 4; ++ct){
    // prefetch all B fragments of this column tile so loads pipeline ahead of WMMA chain
    v2f bfr[8];
    #pragma unroll
    for (int kc = 0; kc < 8; ++kc){
      int k0 = kc*4 + kh;
      bfr[kc].x = Zt[k0][ct*16 + c];
      bfr[kc].y = Zt[k0+1][ct*16 + c];
    }
    v8f acc = {};
    #pragma unroll
    for (int kc = 0; kc < 8; ++kc){
      acc = __builtin_amdgcn_wmma_f32_16x16x4_f32(false, afr[kc], false, bfr[kc],
                                                  (short)0, acc, false, false);
    }
    #pragma unroll
    for (int i = 0; i < 8; ++i){
      int row = i + ((lane < 16) ? 0 : 8);
      int m   = ct*16 + c;
      float sq = x2l[w][row] + z2l[m] - 2.f*acc[i];
      float kx = osg*__expf(sq*nhalf);
      pacc[i] += kx * alphal[aj][m];
    }
  }
  // reduce over m across the 16-lane half
  #pragma unroll
  for (int i = 0; i < 8; ++i){
    float v = pacc[i];
    v += __shfl_xor(v, 1, 32);
    v += __shfl_xor(v, 2, 32);
    v += __shfl_xor(v, 4, 32);
    v += __shfl_xor(v, 8, 32);
    pacc[i] = v;
  }
  if (c == 0){
    int base = (lane < 16) ? 0 : 8;
    #pragma unroll
    for (int i = 0; i < 8; ++i){
      int row = base + i;
      int b = b0 + row;
      float mean = pacc[i];
      float e = eps_proc[(((size_t)t*N_ + n)*DX + g)*B_ + b];
      float xb;
      if (t == 0){
        float e0 = eps_x0[(((size_t)n*DX + 0)*B_ + b)*DX + g];
        xb = qm[b*DX + g] + expf(ql[b*DX + g])*e0;
      } else {
        xb = xm[b*DX + g];
      }
      ws[WS_XBUF + ((size_t)b*N_ + n)*DX + g] = mean + PROC_SD*e + xb;
    }
  }
}

// ---------------- EnKF update per batch b (128 threads) ----------------
__global__ void k_enkf(int t, const float* obs, const float* H, float* ws, float* out){
  __shared__ float Xcl[M_][DX+1];   // rows n (padded to 64), cols d
  __shared__ float Hl[DY][DX];
  __shared__ float Pl[DX][DX+1];
  __shared__ float HPl[DY][DX+1];
  __shared__ float Sl[DY][DY+1];
  __shared__ float Sinvd[DY];
  __shared__ float Kl[DY][DX+1];
  __shared__ float xmean[DX];
  __shared__ float innov[DY];
  int b = blockIdx.x, tid = threadIdx.x;
  const float* Xb = ws + WS_XBUF + (size_t)b*N_*DX;
  for (int i = tid; i < M_*DX; i += 128){
    int nn = i >> 5, d = i & 31;
    Xcl[nn][d] = (nn < N_) ? Xb[nn*DX + d] : 0.f;
  }
  for (int i = tid; i < DY*DX; i += 128) Hl[i >> 5][i & 31] = H[i];
  __syncthreads();
  if (tid < DX){
    float s = 0.f;
    for (int nn = 0; nn < N_; ++nn) s += Xcl[nn][tid];
    xmean[tid] = s / (float)N_;
  }
  __syncthreads();
  for (int i = tid; i < N_*DX; i += 128){
    int nn = i / DX, d = i % DX;
    Xcl[nn][d] -= xmean[d];
  }
  __syncthreads();
  // P = taper .* Xc^T Xc / (N-1)  via WMMA; 4 waves = 4 output tiles
  {
    int w = tid >> 5, lane = tid & 31;
    int rt = w >> 1, ct2 = w & 1;
    int kh = (lane < 16) ? 0 : 2, c = lane & 15;
    v8f acc = {};
    #pragma unroll
    for (int kb = 0; kb < 4; ++kb){           // 4 batches of 4 k-chunks, prefetched
      v2f afr[4], bfr[4];
      #pragma unroll
      for (int kc = 0; kc < 4; ++kc){
        int k0 = (kb*4 + kc)*4 + kh;
        afr[kc].x = Xcl[k0][rt*16 + c];    afr[kc].y = Xcl[k0+1][rt*16 + c];
        bfr[kc].x = Xcl[k0][ct2*16 + c];   bfr[kc].y = Xcl[k0+1][ct2*16 + c];
      }
      #pragma unroll
      for (int kc = 0; kc < 4; ++kc){
        acc = __builtin_amdgcn_wmma_f32_16x16x4_f32(false, afr[kc], false, bfr[kc],
                                                    (short)0, acc, false, false);
      }
    }
    const float invNm1 = 1.f / (float)(N_ - 1);
    #pragma unroll
    for (int i = 0; i < 8; ++i){
      int row = rt*16 + i + ((lane < 16) ? 0 : 8);
      int col = ct2*16 + c;
      Pl[row][col] = ws[WS_TAPER + row*DX + col] * acc[i] * invNm1;
    }
  }
  __syncthreads();
  for (int i = tid; i < DY*DX; i += 128){
    int o = i >> 5, e = i & 31;
    float s = 0.f;
    for (int d = 0; d < DX; ++d) s += Hl[o][d]*Pl[d][e];
    HPl[o][e] = s;
  }
  __syncthreads();
  for (int i = tid; i < DY*DY; i += 128){
    int o = i >> 4, p = i & 15;
    float s = 0.f;
    for (int e = 0; e < DX; ++e) s += HPl[o][e]*Hl[p][e];
    if (o == p) s += EMIS_SD*EMIS_SD;
    Sl[o][p] = s;
  }
  __syncthreads();
  if (tid == 0){  // Cholesky of S (16x16, lower)
    for (int i = 0; i < DY; ++i)
      for (int j = 0; j <= i; ++j){
        float s = Sl[i][j];
        for (int k = 0; k < j; ++k) s -= Sl[i][k]*Sl[j][k];
        Sl[i][j] = (i == j) ? sqrtf(s) : s / Sl[j][j];
      }
  }
  __syncthreads();
  if (tid < DY) Sinvd[tid] = 1.f / Sl[tid][tid];
  __syncthreads();
  if (tid < DX){  // K = S^{-1} HP, column tid
    float y[DY];
    #pragma unroll
    for (int i = 0; i < DY; ++i){
      float s = HPl[i][tid];
      for (int k = 0; k < i; ++k) s -= Sl[i][k]*y[k];
      y[i] = s * Sinvd[i];
    }
    #pragma unroll
    for (int j = DY-1; j >= 0; --j){
      float s = y[j];
      for (int k = j+1; k < DY; ++k) s -= Sl[k][j]*y[k];
      y[j] = s * Sinvd[j];
      Kl[j][tid] = y[j];
    }
  }
  if (tid < DY){
    float s = 0.f;
    for (int d = 0; d < DX; ++d) s += Hl[tid][d]*xmean[d];
    innov[tid] = obs[((size_t)b*T_ + t)*DY + tid] - s;
  }
  __syncthreads();
  if (tid < DX){
    float s = xmean[tid];
    for (int o = 0; o < DY; ++o) s += innov[o]*Kl[o][tid];
    out[1 + ((size_t)t*B_ + b)*DX + tid] = s;
    ws[WS_XM + b*DX + tid] = s;
  }
  if (tid == 0){
    float z[DY]; float ll = 0.f;
    #pragma unroll
    for (int i = 0; i < DY; ++i){
      float s = innov[i];
      for (int k = 0; k < i; ++k) s -= Sl[i][k]*z[k];
      z[i] = s * Sinvd[i];
      ll += -0.5f*z[i]*z[i] - logf(Sl[i][i]);
    }
    ll -= 0.5f*(float)DY*logf(2.f*3.14159265358979323846f);
    ws[WS_LL + t*B_ + b] = ll;
  }
}

// ---------------- final ELBO ----------------
__global__ void k_final(float* ws, float* out){
  __shared__ float red[256];
  int tid = threadIdx.x;
  float s = 0.f;
  for (int i = tid; i < T_*B_; i += 256) s += ws[WS_LL + i];
  red[tid] = s; __syncthreads();
  for (int st = 128; st > 0; st >>= 1){
    if (tid < st) red[tid] += red[tid + st];
    __syncthreads();
  }
  if (tid == 0){
    float lik = red[0] / (float)B_;
    float gp = 0.f;
    for (int g = 0; g < DX; ++g) gp += ws[WS_GPKL + g];
    gp = 0.5f*gp / (float)(T_*B_);
    out[0] = -ws[WS_KL0] - gp + lik;
  }
}

extern "C" void kernel_launch(void* const* d_in, const int* in_sizes, int n_in,
                              void* d_out, int out_size, void* d_ws, size_t ws_size,
                              hipStream_t stream) {
  (void)in_sizes; (void)n_in; (void)out_size; (void)ws_size;
  const float* obs      = (const float*)d_in[0];
  const float* H        = (const float*)d_in[1];
  const float* Z        = (const float*)d_in[2];
  const float* m_u      = (const float*)d_in[3];
  const float* Ls       = (const float*)d_in[4];
  const float* log_ls   = (const float*)d_in[5];
  const float* log_os   = (const float*)d_in[6];
  const float* qm       = (const float*)d_in[7];
  const float* ql       = (const float*)d_in[8];
  const float* eps_x0   = (const float*)d_in[9];
  const float* eps_U    = (const float*)d_in[10];
  const float* eps_proc = (const float*)d_in[11];
  // d_in[12] (noise_y_all) is dead in the reference (X_post unused).
  float* ws  = (float*)d_ws;
  float* out = (float*)d_out;

  k_taper <<<dim3(4),  dim3(256), 0, stream>>>(ws);
  k_kl0   <<<dim3(1),  dim3(256), 0, stream>>>(qm, ql, ws);
  k_kzz   <<<dim3(DX), dim3(M_),  0, stream>>>(Z, log_ls, log_os, ws);
  k_ualpha<<<dim3(DX), dim3(M_),  0, stream>>>(Ls, m_u, eps_U, ws);
  k_gpkl  <<<dim3(DX), dim3(M_),  0, stream>>>(Ls, m_u, ws);
  for (int t = 0; t < T_; ++t){
    k_predict<<<dim3(25, DX), dim3(128), 0, stream>>>(t, Z, log_ls, log_os, qm, ql,
                                                      eps_x0, eps_proc, ws);
    k_enkf   <<<dim3(B_),     dim3(128), 0, stream>>>(t, obs, H, ws, out);
  }
  k_final<<<dim3(1), dim3(256), 0, stream>>>(ws, out);
}